// Dynamic_sampling_86526411145607
// MI455X (gfx1250) — compile-verified
//
#include <hip/hip_runtime.h>

#define B_  4
#define C_  5
#define P_  8192
#define S_  4096
#define K_  16

typedef __attribute__((ext_vector_type(2))) float v2f;
typedef __attribute__((ext_vector_type(8))) float v8f;

// --------------------------------------------------------------------------
// Kernel 1: build the WMMA B-operand in half-split planes:
//   plane0[p] = (qx, qy)          (K=0,1  -> lanes 0-15)
//   plane1[p] = (qz, -||q||^2)    (K=2,3  -> lanes 16-31)
// Folding -qq into the K=3 slot lets one v_wmma_f32_16x16x4_f32 produce the
// per-tile score 2*(r.q) - qq  (the -rr term is a per-row constant and cannot
// change per-row top-k ranking, so it is dropped).  The split-plane layout
// means each lane does a single global_load_b64 with no component selects.
// --------------------------------------------------------------------------
__global__ __launch_bounds__(256) void qprep_kernel(const float* __restrict__ x,
                                                    float2* __restrict__ qsw) {
  int gid = blockIdx.x * 256 + threadIdx.x;
  if (gid >= B_ * P_) return;
  int b = gid >> 13;          // / P_
  int p = gid & (P_ - 1);
  const float* xb = x + (size_t)b * C_ * P_;
  float qx = xb[p], qy = xb[P_ + p], qz = xb[2 * P_ + p];
  float2* qb = qsw + (size_t)b * 2 * P_;
  qb[p]       = make_float2(qx, qy);
  qb[P_ + p]  = make_float2(qz, -(qx * qx + qy * qy + qz * qz));
}

// --------------------------------------------------------------------------
// Kernel 2: per-batch bitonic argsort of 8192 (key,idx) pairs in LDS (64 KB,
// well under the 320 KB/WGP budget).  Emits the A-operand in the same
// half-split planes: plane0[s] = (2rx, 2ry), plane1[s] = (2rz, 1).
// --------------------------------------------------------------------------
__global__ __launch_bounds__(1024) void sort_kernel(const float* __restrict__ rand_num,
                                                    const float* __restrict__ x,
                                                    float2* __restrict__ rsw) {
  __shared__ float key[P_];
  __shared__ int   ind[P_];
  const int b = blockIdx.x;
  const float* rn = rand_num + (size_t)b * P_;
  for (int i = threadIdx.x; i < P_; i += 1024) { key[i] = rn[i]; ind[i] = i; }
  __syncthreads();
  for (int k = 2; k <= P_; k <<= 1) {
    for (int j = k >> 1; j > 0; j >>= 1) {
      for (int t = threadIdx.x; t < P_ / 2; t += 1024) {
        int i   = ((t & ~(j - 1)) << 1) | (t & (j - 1));
        int ixj = i | j;
        bool up = ((i & k) == 0);
        float a = key[i], c = key[ixj];
        bool sw = up ? (a > c) : (a < c);
        if (sw) {
          int ia = ind[i], ic = ind[ixj];
          key[i] = c; key[ixj] = a;
          ind[i] = ic; ind[ixj] = ia;
        }
      }
      __syncthreads();
    }
  }
  const float* xb = x + (size_t)b * C_ * P_;
  float2* rb = rsw + (size_t)b * 2 * S_;
  for (int s = threadIdx.x; s < S_; s += 1024) {
    int pi = ind[s];
    rb[s]      = make_float2(2.f * xb[pi], 2.f * xb[P_ + pi]);
    rb[S_ + s] = make_float2(2.f * xb[2 * P_ + pi], 1.0f);
  }
}

// --------------------------------------------------------------------------
// Kernel 3: one wave32 per 16 sample rows.  Sweep all 8192 candidates in
// 32-column super-tiles: two v_wmma_f32_16x16x4_f32 (shared A operand) per
// iteration, one barrier pair.  D tiles bounce through padded LDS rows
// (80 B stride, 16-B aligned) so the readback is two ds_load_b128 per lane.
// Each lane owns (row, half) and keeps a register-resident sorted top-16
// (constant-indexed -> no scratch).  Final cross-half merge + gather.
// --------------------------------------------------------------------------
__global__ __launch_bounds__(32) void knn_gather_kernel(const float* __restrict__ x,
                                                        const float2* __restrict__ rsw,
                                                        const float2* __restrict__ qsw,
                                                        float* __restrict__ out) {
  __shared__ __align__(16) float tile[2][16][20];
  __shared__ float mv[32][16];
  __shared__ int   mi[32][16];
  __shared__ int   fidx[16][16];

  const int wg   = blockIdx.x;
  const int b    = wg / (S_ / 16);
  const int srow = (wg % (S_ / 16)) * 16;
  const int lane = threadIdx.x;
  const int m    = lane & 15;
  const int half = lane >> 4;

  // A operand (16x4 f32): lanes 0-15 hold K=0,1 ; lanes 16-31 hold K=2,3.
  const float2* rh = rsw + (size_t)b * 2 * S_ + (size_t)half * S_;
  const float2* qh = qsw + (size_t)b * 2 * P_ + (size_t)half * P_;
  const v2f aop = *(const v2f*)&rh[srow + m];

  float tv[16];
  int   ti[16];
#pragma unroll
  for (int i = 0; i < 16; ++i) { tv[i] = -3.4e38f; ti[i] = 0; }

  for (int pt = 0; pt < P_; pt += 32) {
    // B operand (4x16 f32): one global_load_b64 per lane, no selects.
    const v2f b0 = *(const v2f*)&qh[pt + m];
    const v2f b1 = *(const v2f*)&qh[pt + 16 + m];
    if (pt + 32 < P_) __builtin_prefetch(&qh[pt + 32 + m], 0, 1);

    v8f acc0 = {}, acc1 = {};
    acc0 = __builtin_amdgcn_wmma_f32_16x16x4_f32(false, aop, false, b0,
                                                 (short)0, acc0, false, false);
    acc1 = __builtin_amdgcn_wmma_f32_16x16x4_f32(false, aop, false, b1,
                                                 (short)0, acc1, false, false);
    // D layout: VGPR v holds row (v + 8*half), column m.
#pragma unroll
    for (int v = 0; v < 8; ++v) {
      tile[0][v + 8 * half][m] = acc0[v];
      tile[1][v + 8 * half][m] = acc1[v];
    }
    __syncthreads();

    // Lane (m, half) owns row m, columns half*8 .. half*8+7 of each sub-tile.
#pragma unroll
    for (int t = 0; t < 2; ++t) {
      const float4 lo = *(const float4*)&tile[t][m][half * 8];
      const float4 hi = *(const float4*)&tile[t][m][half * 8 + 4];
      const float vals[8] = {lo.x, lo.y, lo.z, lo.w, hi.x, hi.y, hi.z, hi.w};
#pragma unroll
      for (int j = 0; j < 8; ++j) {
        float val  = vals[j];
        int   pidx = pt + t * 16 + half * 8 + j;
        if (val > tv[15]) {        // strict >  ==> lower index wins ties
          tv[15] = val; ti[15] = pidx;
#pragma unroll
          for (int i = 15; i > 0; --i) {
            if (tv[i] > tv[i - 1]) {
              float tf = tv[i]; tv[i] = tv[i - 1]; tv[i - 1] = tf;
              int   tn = ti[i]; ti[i] = ti[i - 1]; ti[i - 1] = tn;
            }
          }
        }
      }
    }
    __syncthreads();
  }

  // Publish per-lane sorted lists.
#pragma unroll
  for (int j = 0; j < 16; ++j) { mv[lane][j] = tv[j]; mi[lane][j] = ti[j]; }
  __syncthreads();

  // Merge the two halves of each row (descending value, ascending index tie).
  if (half == 0) {
    int ia = 0, ib = 0;
#pragma unroll
    for (int j = 0; j < 16; ++j) {
      float va = mv[m][ia];      int na = mi[m][ia];
      float vb = mv[m + 16][ib]; int nb = mi[m + 16][ib];
      bool takeA = (va > vb) || (va == vb && na < nb);
      fidx[m][j] = takeA ? na : nb;
      ia += takeA ? 1 : 0;
      ib += takeA ? 0 : 1;
    }
  }
  __syncthreads();

  // Gather: out[b][c][srow+sl][k] = x[b][c][ fidx[sl][k] ]
  const float* xb = x + (size_t)b * C_ * P_;
  float* ob = out + (size_t)b * C_ * S_ * K_;
  for (int e = lane; e < 16 * C_ * K_; e += 32) {
    int kk = e & 15;
    int t  = e >> 4;
    int sl = t & 15;
    int cc = t >> 4;            // 0..4
    int pidx = fidx[sl][kk];
    ob[((size_t)cc * S_ + (srow + sl)) * K_ + kk] = xb[(size_t)cc * P_ + pidx];
  }
}

// --------------------------------------------------------------------------
extern "C" void kernel_launch(void* const* d_in, const int* in_sizes, int n_in,
                              void* d_out, int out_size, void* d_ws, size_t ws_size,
                              hipStream_t stream) {
  (void)in_sizes; (void)n_in; (void)out_size; (void)ws_size;
  const float* x  = (const float*)d_in[0];
  const float* rn = (const float*)d_in[1];
  // d_in[2] is s_num == 4096, fixed by the reference setup.
  float* out = (float*)d_out;

  float2* rsw = (float2*)d_ws;                                          // 256 KiB
  float2* qsw = (float2*)((char*)d_ws + (size_t)B_ * 2 * S_ * sizeof(float2)); // +512 KiB

  hipLaunchKernelGGL(qprep_kernel, dim3((B_ * P_) / 256), dim3(256), 0, stream,
                     x, qsw);
  hipLaunchKernelGGL(sort_kernel, dim3(B_), dim3(1024), 0, stream, rn, x, rsw);
  hipLaunchKernelGGL(knn_gather_kernel, dim3(B_ * (S_ / 16)), dim3(32), 0, stream,
                     x, rsw, qsw, out);
}